// PSA_p_29850022707958
// MI455X (gfx1250) — compile-verified
//
#include <hip/hip_runtime.h>
#include <hip/hip_bf16.h>

// PSA (polarized self-attention) fused pipeline for MI455X (gfx1250, wave32).
// B=16, C=512, H=W=64, inter=256, R=8. fp32 throughout; big GEMMs via
// V_WMMA_F32_16X16X4_F32 (full-precision fp32 WMMA).
//
// v2: K-pair-interleaved LDS layout (B fragment = single aligned ds_load_b64)
//     + double-buffered x staging so global loads overlap the WMMA stage.

typedef __attribute__((ext_vector_type(2))) float v2f;
typedef __attribute__((ext_vector_type(8))) float v8f;

#define C_DIM   512
#define INTER   256
#define H_DIM   64
#define W_DIM   64
#define HW      4096
#define PH      8
#define PW      8
#define NREGPX  64      // pixels per 8x8 region
#define KCHUNK  64      // channels staged in LDS per iteration
#define LXSTRIDE 160    // dword row stride for k-pair buffer (128 used + 32 pad)

// ---------------------------------------------------------------------------
// Kernel 0: zero the channel-sum accumulator (workspace is poisoned 0xAA).
// ---------------------------------------------------------------------------
__global__ void k_zero(float* p, int n) {
    int i = blockIdx.x * 256 + threadIdx.x;
    if (i < n) p[i] = 0.0f;
}

// ---------------------------------------------------------------------------
// Kernel 1: per-region fused spatial pool.
//   - v     = Wv_r * x_region   (WMMA, kept on-chip in LDS)
//   - theta = Wv_l * x_region   (WMMA, streamed to global for channel pool)
//   - q     = Wq_r * x_region, softmax over the 64 region pixels
//   - ctx   = v * softmax(q);  mask_ch = sigmoid(Wup * ctx)  -> global
//   - sum_x[b][c] += region channel sums (atomic, for fused avg-pool)
// grid = 1024 (B*PH*PW), block = 256 (8 waves).
// ---------------------------------------------------------------------------
__global__ __launch_bounds__(256) void k_region(
    const float* __restrict__ x,
    const float* __restrict__ wqr,   // (1,512)
    const float* __restrict__ wvr,   // (256,512)
    const float* __restrict__ wup,   // (512,256)
    const float* __restrict__ wvl,   // (256,512)
    float* __restrict__ theta,       // (B,256,4096)
    float* __restrict__ sumx,        // (B,512)
    float* __restrict__ maskch)      // (B,PH,PW,512)
{
    // x chunk, double buffered, K-pair interleaved:
    //   element (k, p) -> lxp[buf][k>>1][p*2 + (k&1)]
    __shared__ float lxp[2][KCHUNK / 2][LXSTRIDE];
    __shared__ float lv[INTER][NREGPX + 1];    // v tiles: 256 x 64 (+pad)
    __shared__ float lq[NREGPX];
    __shared__ float lctx[INTER];
    __shared__ float lsc[1];

    const int blk  = blockIdx.x;
    const int b    = blk >> 6;
    const int preg = (blk >> 3) & 7;
    const int qreg = blk & 7;
    const int h0   = preg * 8;
    const int w0   = qreg * 8;
    const int t    = threadIdx.x;
    const int lane = t & 31;
    const int wave = t >> 5;

    v8f zero = {};
    v8f accv[2][4], acct[2][4];
#pragma unroll
    for (int rt = 0; rt < 2; ++rt)
#pragma unroll
        for (int ct = 0; ct < 4; ++ct) { accv[rt][ct] = zero; acct[rt][ct] = zero; }

    float qacc = 0.0f;
    const int mrow0   = wave * 32;          // this wave owns output rows [mrow0, mrow0+32)
    const int halfsel = (lane >> 4) << 1;   // 0 for lanes 0-15, 2 for lanes 16-31
    const int lan15   = lane & 15;

    // staging assignment: channel pair cp = t>>3 (0..31), row oct = t&7
    const int cp  = t >> 3;
    const int oct = t & 7;
    float4 st0a, st0b, st1a, st1b;

    auto load_stage = [&](int chunk) {
        const float* s0 = x + (((size_t)(b * C_DIM + chunk * KCHUNK + 2 * cp)) * H_DIM
                               + (h0 + oct)) * W_DIM + w0;
        const float* s1 = s0 + (size_t)H_DIM * W_DIM;   // next channel of the pair
        st0a = *(const float4*)(s0);
        st0b = *(const float4*)(s0 + 4);
        st1a = *(const float4*)(s1);
        st1b = *(const float4*)(s1 + 4);
    };
    auto store_stage = [&](int buf) {
        float2* dst = (float2*)&lxp[buf][cp][oct * 16];
        dst[0] = make_float2(st0a.x, st1a.x);
        dst[1] = make_float2(st0a.y, st1a.y);
        dst[2] = make_float2(st0a.z, st1a.z);
        dst[3] = make_float2(st0a.w, st1a.w);
        dst[4] = make_float2(st0b.x, st1b.x);
        dst[5] = make_float2(st0b.y, st1b.y);
        dst[6] = make_float2(st0b.z, st1b.z);
        dst[7] = make_float2(st0b.w, st1b.w);
    };

    // prologue: stage chunk 0 into buffer 0
    load_stage(0);
    store_stage(0);
    int nb = 0;

    for (int chunk = 0; chunk < 8; ++chunk) {
        __syncthreads();                 // buf[nb] ready; prior reads of other buf done
        const int cb    = nb;
        const int kbase = chunk * KCHUNK;

        // ---- issue next chunk's global loads early (covered by WMMA below)
        if (chunk < 7) load_stage(chunk + 1);

        // ---- scalar side work on waves 0-3 (EXEC restored before WMMA)
        if (t < 64) {                    // query: 1-channel conv partial
            float s = 0.0f;
            for (int kp = 0; kp < KCHUNK / 2; ++kp) {
                const float2 vp = *(const float2*)&lxp[cb][kp][t * 2];
                s += wqr[kbase + 2 * kp] * vp.x + wqr[kbase + 2 * kp + 1] * vp.y;
            }
            qacc += s;
        } else if (t < 128) {            // per-channel region sum -> global atomic
            const int ch   = t - 64;
            const int half = ch & 1;
            const int row  = ch >> 1;
            float s = 0.0f;
            for (int p = 0; p < NREGPX; ++p) {
                const int pe = (p + ch) & 63;       // skew to avoid bank conflicts
                s += lxp[cb][row][pe * 2 + half];
            }
            atomicAdd(&sumx[b * C_DIM + kbase + ch], s);
        }

        // ---- fp32 WMMA: v += Wvr * X,  theta += Wvl * X   (K in steps of 4)
#pragma unroll
        for (int ks = 0; ks < 16; ++ks) {
            const int kk = kbase + ks * 4;          // global channel of this K-step
            v2f afv[2], aft[2];
#pragma unroll
            for (int rt = 0; rt < 2; ++rt) {
                const int M  = mrow0 + rt * 16 + lan15;
                const int ko = kk + halfsel;        // A 16x4: lanes0-15 K={0,1}, lanes16-31 K={2,3}
                v2f av; av.x = wvr[(size_t)M * C_DIM + ko]; av.y = wvr[(size_t)M * C_DIM + ko + 1];
                v2f at; at.x = wvl[(size_t)M * C_DIM + ko]; at.y = wvl[(size_t)M * C_DIM + ko + 1];
                afv[rt] = av; aft[rt] = at;
            }
            const int kprow = ks * 2 + (lane >> 4); // lanes0-15 pair {K,K+1}, lanes16-31 {K+2,K+3}
#pragma unroll
            for (int ct = 0; ct < 4; ++ct) {
                const v2f bf = *(const v2f*)&lxp[cb][kprow][(ct * 16 + lan15) * 2];
#pragma unroll
                for (int rt = 0; rt < 2; ++rt) {
                    accv[rt][ct] = __builtin_amdgcn_wmma_f32_16x16x4_f32(
                        false, afv[rt], false, bf, (short)0, accv[rt][ct], false, false);
                    acct[rt][ct] = __builtin_amdgcn_wmma_f32_16x16x4_f32(
                        false, aft[rt], false, bf, (short)0, acct[rt][ct], false, false);
                }
            }
        }

        // ---- park next chunk in the idle buffer
        if (chunk < 7) { store_stage(1 - nb); nb = 1 - nb; }
    }

    // ---- unload tiles: v -> LDS, theta -> global (C/D: VGPR r holds M=r / M=r+8)
#pragma unroll
    for (int rt = 0; rt < 2; ++rt) {
        const int Mb = mrow0 + rt * 16;
#pragma unroll
        for (int ct = 0; ct < 4; ++ct) {
            const int col = ct * 16 + lan15;            // region pixel index
            const int n   = (h0 + (col >> 3)) * W_DIM + w0 + (col & 7);
            const int rb  = Mb + ((lane >> 4) << 3);    // +8 for upper half-wave
#pragma unroll
            for (int vg = 0; vg < 8; ++vg) {
                const int row = rb + vg;
                lv[row][col] = accv[rt][ct][vg];
                theta[((size_t)b * INTER + row) * HW + n] = acct[rt][ct][vg];
            }
        }
    }
    if (t < 64) lq[t] = qacc;
    __syncthreads();

    // ---- softmax over the 64 region pixels (tiny; single thread)
    if (t == 0) {
        float m = -1e30f;
        for (int p = 0; p < NREGPX; ++p) m = fmaxf(m, lq[p]);
        float s = 0.0f;
        for (int p = 0; p < NREGPX; ++p) { float e = __expf(lq[p] - m); lq[p] = e; s += e; }
        lsc[0] = 1.0f / s;
    }
    __syncthreads();

    // ---- ctx[i] = sum_p v[i][p] * softmax(q)[p]   (256 threads, 64 MACs each)
    {
        const float rinv = lsc[0];
        float cv = 0.0f;
        for (int p = 0; p < NREGPX; ++p) cv += lv[t][p] * lq[p];
        lctx[t] = cv * rinv;
    }
    __syncthreads();

    // ---- up = Wup * ctx; mask_ch = sigmoid(up)
    float* mch = maskch + (size_t)blk * C_DIM;   // blk == (b*8+p)*8+q
    for (int o = t; o < C_DIM; o += 256) {
        const float* wrow = wup + (size_t)o * INTER;
        float u = 0.0f;
        for (int i = 0; i < INTER; ++i) u += wrow[i] * lctx[i];
        mch[o] = 1.0f / (1.0f + __expf(-u));
    }
}

// ---------------------------------------------------------------------------
// Kernel 2: avg = (Wq_l * sum_x) / HW  ->  (B, 4, INTER) flat as (B,1024)
// ---------------------------------------------------------------------------
__global__ void k_avg(const float* __restrict__ wql, const float* __restrict__ sumx,
                      float* __restrict__ avg)
{
    const int b = blockIdx.x;
    const int t = threadIdx.x;
    const float* sx = sumx + b * C_DIM;
    for (int r = 0; r < 4; ++r) {
        const int o = r * 256 + t;
        const float* wr = wql + (size_t)o * C_DIM;
        float s = 0.0f;
        for (int c = 0; c < C_DIM; ++c) s += wr[c] * sx[c];
        avg[b * 1024 + o] = s * (1.0f / (float)HW);
    }
}

// ---------------------------------------------------------------------------
// Kernel 3: ctx2[b,k,n] = sum_i avg[b,k,i] * theta[b,i,n]
// grid = B * 16 n-tiles of 256; coalesced theta reads.
// ---------------------------------------------------------------------------
__global__ void k_ctx2(const float* __restrict__ avg, const float* __restrict__ theta,
                       float* __restrict__ ctx2)
{
    const int blk  = blockIdx.x;
    const int b    = blk >> 4;
    const int tile = blk & 15;
    const int n    = tile * 256 + threadIdx.x;
    const float* av = avg + b * 1024;
    const float* th = theta + (size_t)b * INTER * HW + n;
    float a0 = 0, a1 = 0, a2 = 0, a3 = 0;
    for (int i = 0; i < INTER; ++i) {
        const float tv = th[(size_t)i * HW];
        a0 += av[i]       * tv;
        a1 += av[256 + i] * tv;
        a2 += av[512 + i] * tv;
        a3 += av[768 + i] * tv;
    }
    const size_t base = ((size_t)b * 4) * HW + n;
    ctx2[base]          = a0;
    ctx2[base + HW]     = a1;
    ctx2[base + 2 * HW] = a2;
    ctx2[base + 3 * HW] = a3;
}

// ---------------------------------------------------------------------------
// Kernel 4: row softmax over 4096 + sigmoid -> mask_sp.  grid = B*4.
// ---------------------------------------------------------------------------
__global__ __launch_bounds__(256) void k_softmax_sig(const float* __restrict__ ctx2,
                                                     float* __restrict__ masksp)
{
    __shared__ float red[256];
    __shared__ float sc[2];
    const int row = blockIdx.x;
    const int t   = threadIdx.x;
    const float* src = ctx2 + (size_t)row * HW;

    float m = -1e30f;
    for (int n = t; n < HW; n += 256) m = fmaxf(m, src[n]);
    red[t] = m; __syncthreads();
    for (int s = 128; s > 0; s >>= 1) { if (t < s) red[t] = fmaxf(red[t], red[t + s]); __syncthreads(); }
    if (t == 0) sc[0] = red[0];
    __syncthreads();
    m = sc[0];

    float s = 0.0f;
    for (int n = t; n < HW; n += 256) s += __expf(src[n] - m);
    red[t] = s; __syncthreads();
    for (int k = 128; k > 0; k >>= 1) { if (t < k) red[t] += red[t + k]; __syncthreads(); }
    if (t == 0) sc[1] = 1.0f / red[0];
    __syncthreads();
    const float inv = sc[1];

    float* dst = masksp + (size_t)row * HW;
    for (int n = t; n < HW; n += 256) {
        const float e = __expf(src[n] - m) * inv;
        dst[n] = 1.0f / (1.0f + __expf(-e));
    }
}

// ---------------------------------------------------------------------------
// Kernel 5: out = x * (1 + mask_sp + mask_ch)
// ---------------------------------------------------------------------------
__global__ void k_final(const float* __restrict__ x, const float* __restrict__ masksp,
                        const float* __restrict__ maskch, float* __restrict__ out)
{
    const size_t idx = (size_t)blockIdx.x * 256 + threadIdx.x;
    const int n = (int)(idx & 4095);
    const int c = (int)((idx >> 12) & 511);
    const int b = (int)(idx >> 21);
    const int h = n >> 6, w = n & 63;
    const float m1 = masksp[((size_t)(b * 4 + (c >> 7))) * HW + n];
    const float m2 = maskch[((size_t)((b * 8 + (h >> 3)) * 8 + (w >> 3))) * C_DIM + c];
    const float xv = x[idx];
    out[idx] = xv * (1.0f + m1 + m2);
}

// ---------------------------------------------------------------------------
extern "C" void kernel_launch(void* const* d_in, const int* in_sizes, int n_in,
                              void* d_out, int out_size, void* d_ws, size_t ws_size,
                              hipStream_t stream) {
    const float* x   = (const float*)d_in[0];
    const float* wqr = (const float*)d_in[1];
    const float* wvr = (const float*)d_in[2];
    const float* wup = (const float*)d_in[3];
    const float* wql = (const float*)d_in[4];
    const float* wvl = (const float*)d_in[5];
    float* out = (float*)d_out;

    // workspace layout (floats)
    float* ws     = (float*)d_ws;
    float* theta  = ws;                                 // 16*256*4096 = 16,777,216
    float* sumx   = theta  + (size_t)16 * INTER * HW;   // 16*512     = 8,192
    float* avg    = sumx   + 16 * C_DIM;                // 16*1024    = 16,384
    float* ctx2   = avg    + 16 * 1024;                 // 16*4*4096  = 262,144
    float* masksp = ctx2   + (size_t)16 * 4 * HW;       // 16*4*4096  = 262,144
    float* maskch = masksp + (size_t)16 * 4 * HW;       // 16*64*512  = 524,288

    k_zero<<<32, 256, 0, stream>>>(sumx, 16 * C_DIM);
    k_region<<<16 * PH * PW, 256, 0, stream>>>(x, wqr, wvr, wup, wvl, theta, sumx, maskch);
    k_avg<<<16, 256, 0, stream>>>(wql, sumx, avg);
    k_ctx2<<<16 * 16, 256, 0, stream>>>(avg, theta, ctx2);
    k_softmax_sig<<<16 * 4, 256, 0, stream>>>(ctx2, masksp);
    k_final<<<(16 * C_DIM * HW) / 256, 256, 0, stream>>>(x, masksp, maskch, out);
}